// DotAtt_10179072492025
// MI455X (gfx1250) — compile-verified
//
#include <hip/hip_runtime.h>
#include <stdint.h>

typedef __attribute__((ext_vector_type(2))) float v2f;
typedef __attribute__((ext_vector_type(4))) float v4f;
typedef __attribute__((ext_vector_type(8))) float v8f;

constexpr int B_  = 16;
constexpr int LQ  = 2048;
constexpr int LK  = 2048;
constexpr int D_  = 64;
constexpr float SCALE    = 64.0f;
constexpr float INV_KEEP = 1.0f / 0.9f;
constexpr uint32_t DROP_THRESH = 429496730u;   // ~0.1 * 2^32
constexpr int LPAD = 20;                       // LDS row pitch (dwords), 16B-aligned

__device__ __forceinline__ v4f ld4(const float* p) {
  return *reinterpret_cast<const v4f*>(p);
}

// Deterministic hash-based Bernoulli(keep=0.9) dropout mask (seed 42 mixed in).
__device__ __forceinline__ float keep_mask(uint32_t idx) {
  uint32_t x = idx ^ 0x2545F491u;
  x *= 0x9E3779B9u;
  x ^= x >> 16; x *= 0x7FEB352Du;
  x ^= x >> 15; x *= 0x846CA68Bu;
  x ^= x >> 16;
  return (x >= DROP_THRESH) ? 1.0f : 0.0f;
}

// One lane's share of a 16x64 operand tile: 32 contiguous floats at
// row + 32*hi, loaded as 8 x b128. K-groups are remapped so WMMA chunk c
// covers K = {2c, 2c+1} (lanes 0-15) and {32+2c, 32+2c+1} (lanes 16-31);
// the dot product is K-order invariant as long as A and B agree.
struct Tile8 { v4f f[8]; };

__device__ __forceinline__ Tile8 load_tile(const float* lane_base) {
  Tile8 t;
#pragma unroll
  for (int i = 0; i < 8; ++i) t.f[i] = ld4(lane_base + 4 * i);
  return t;
}

// Operand for WMMA chunk c (0..15): lane-local floats {2c, 2c+1}.
__device__ __forceinline__ v2f pick(const Tile8& t, int c) {
  v2f a;
  a[0] = t.f[c >> 1][(c & 1) * 2 + 0];
  a[1] = t.f[c >> 1][(c & 1) * 2 + 1];
  return a;
}

// S = Q K^T over D=64 as two independent 8-deep WMMA chains.
__device__ __forceinline__ v8f score_tile(const Tile8& qa, const Tile8& kb) {
  v8f acc0 = {}, acc1 = {};
#pragma unroll
  for (int c = 0; c < 8; ++c) {
    acc0 = __builtin_amdgcn_wmma_f32_16x16x4_f32(false, pick(qa, 2 * c), false,
                                                 pick(kb, 2 * c), (short)0,
                                                 acc0, false, false);
    acc1 = __builtin_amdgcn_wmma_f32_16x16x4_f32(false, pick(qa, 2 * c + 1), false,
                                                 pick(kb, 2 * c + 1), (short)0,
                                                 acc1, false, false);
  }
  return acc0 + acc1;
}

// ---------------------------------------------------------------------------
// Pass 1: per-(b,k) column softmax stats over the query axis.
// One wave owns 16 k-columns; K tile register-resident; stream 128 q-tiles
// with explicit double buffering.
// ---------------------------------------------------------------------------
__global__ __launch_bounds__(256, 2)
void col_stats_kernel(const float* __restrict__ Qm, const float* __restrict__ Km,
                      float* __restrict__ colmax, float* __restrict__ colsum) {
  const int lane  = threadIdx.x & 31;
  const int wave  = threadIdx.x >> 5;
  const int tpb   = LK / 128;
  const int b     = blockIdx.x / tpb;
  const int kbase = ((blockIdx.x % tpb) * 8 + wave) * 16;

  const int row16 = lane & 15;
  const int hi    = lane >> 4;

  const Tile8 kreg =
      load_tile(Km + ((size_t)b * LK + kbase + row16) * D_ + 32 * hi);

  const float* qptr = Qm + ((size_t)b * LQ + row16) * D_ + 32 * hi;

  float m = -INFINITY;
  float z = 0.0f;

  Tile8 cur = load_tile(qptr);
#pragma unroll 2
  for (int qt = 0; qt < LQ / 16; ++qt) {
    const float* nptr = qptr + 16 * D_;
    __builtin_prefetch(nptr + 16 * D_, 0, 1);
    const Tile8 nxt = load_tile(qt < LQ / 16 - 1 ? nptr : qptr);

    const v8f acc = score_tile(cur, kreg);

    // lane holds column N=row16, q rows = qt*16 + r + 8*hi; online (m, Z).
    float s[8];
    float tm = -INFINITY;
#pragma unroll
    for (int r = 0; r < 8; ++r) { s[r] = acc[r] * SCALE; tm = fmaxf(tm, s[r]); }
    const float nm = fmaxf(m, tm);
    float ts = 0.0f;
#pragma unroll
    for (int r = 0; r < 8; ++r) ts += __expf(s[r] - nm);
    z = z * __expf(m - nm) + ts;
    m = nm;

    cur = nxt;
    qptr = nptr;
  }

  // Lane pair (L, L^16) covers the same column: merge the two halves.
  const float om = __shfl_xor(m, 16, 32);
  const float oz = __shfl_xor(z, 16, 32);
  const float nm = fmaxf(m, om);
  const float Z  = z * __expf(m - nm) + oz * __expf(om - nm);
  if (lane < 16) {
    colmax[(size_t)b * LK + kbase + lane] = nm;
    colsum[(size_t)b * LK + kbase + lane] = Z;
  }
}

// ---------------------------------------------------------------------------
// Pass 2: O = dropout(softmax_q(64*QK^T)) * V, fused (scores never hit HBM).
// One wave owns a 16-row q tile (Q register-resident); streams 128 k-tiles
// (double-buffered). W tile transposed C-layout -> A-layout via padded LDS.
// ---------------------------------------------------------------------------
__global__ __launch_bounds__(256, 2)
void attn_out_kernel(const float* __restrict__ Qm, const float* __restrict__ Km,
                     const float* __restrict__ Vm,
                     const float* __restrict__ colmax,
                     const float* __restrict__ colsum,
                     float* __restrict__ Om) {
  __shared__ __align__(16) float wlds[8][16 * LPAD];
  const int lane  = threadIdx.x & 31;
  const int wave  = threadIdx.x >> 5;
  const int tpb   = LQ / 128;
  const int b     = blockIdx.x / tpb;
  const int qbase = ((blockIdx.x % tpb) * 8 + wave) * 16;

  const int row16 = lane & 15;
  const int hi    = lane >> 4;

  const Tile8 qreg =
      load_tile(Qm + ((size_t)b * LQ + qbase + row16) * D_ + 32 * hi);

  v8f o[4] = {{}, {}, {}, {}};                   // 16q x 64d fp32 accum

  const float* kptr = Km + ((size_t)b * LK + row16) * D_ + 32 * hi;
  Tile8 kcur = load_tile(kptr);

#pragma unroll 2
  for (int kt = 0; kt < LK / 16; ++kt) {
    const int kbase = kt * 16;
    const float* knext = kptr + 16 * D_;
    __builtin_prefetch(knext + 16 * D_, 0, 1);
    const Tile8 knxt = load_tile(kt < LK / 16 - 1 ? knext : kptr);

    // ---- V operands (B = V[k,d]) + column stats, issued before the score
    //      chain so their latency hides under 16 WMMAs + exp work ---------
    const float* vbase = Vm + ((size_t)b * LK + kbase) * D_ + row16;
    __builtin_prefetch(vbase + 16 * D_ + 8 * hi * D_, 0, 1);
    v2f bv[4][4];                                // [k-chunk][d-tile]
#pragma unroll
    for (int c2 = 0; c2 < 4; ++c2) {
      // remapped k-groups: chunk c2 -> rows {2c2, 2c2+1} (lo) / {8+2c2, ...} (hi)
      const float* v0 = vbase + (2 * c2 + 8 * hi) * D_;
#pragma unroll
      for (int t = 0; t < 4; ++t) {
        v2f tmp;
        tmp[0] = v0[t * 16];
        tmp[1] = v0[D_ + t * 16];
        bv[c2][t] = tmp;
      }
    }
    const size_t col = (size_t)b * LK + kbase + row16;
    const float mc   = colmax[col];
    const float inv  = INV_KEEP / colsum[col];

    // ---- score tile: S = 64 * Q K^T -------------------------------------
    const v8f acc = score_tile(qreg, kcur);

    // ---- normalize + dropout, stage into LDS for the transpose ----------
#pragma unroll
    for (int r = 0; r < 8; ++r) {
      const int q = qbase + r + 8 * hi;
      const uint32_t idx =
          (uint32_t)(b * LQ + q) * (uint32_t)LK + (uint32_t)(kbase + row16);
      const float w = __expf(acc[r] * SCALE - mc) * inv * keep_mask(idx);
      wlds[wave][(r + 8 * hi) * LPAD + row16] = w;  // W[q_local][k_local]
    }
    // Same-wave LDS ops are in-order (DScnt) -> no barrier needed.

    // A = W in A-layout: lane-local k floats [8*hi, 8*hi+8) -> 2 x ds_load_b128
    v4f w4[2];
    w4[0] = ld4(&wlds[wave][row16 * LPAD + 8 * hi + 0]);
    w4[1] = ld4(&wlds[wave][row16 * LPAD + 8 * hi + 4]);

    // ---- O += W * V  (4 independent accumulator chains over d-tiles) ----
#pragma unroll
    for (int c2 = 0; c2 < 4; ++c2) {
      v2f aw;
      aw[0] = w4[c2 >> 1][(c2 & 1) * 2 + 0];
      aw[1] = w4[c2 >> 1][(c2 & 1) * 2 + 1];
#pragma unroll
      for (int t = 0; t < 4; ++t) {
        o[t] = __builtin_amdgcn_wmma_f32_16x16x4_f32(false, aw, false,
                                                     bv[c2][t], (short)0,
                                                     o[t], false, false);
      }
    }

    kcur = knxt;
    kptr = knext;
  }

  // ---- write 16q x 64d output tile (lanes contiguous in d) --------------
  float* orow = Om + ((size_t)b * LQ + qbase) * D_;
#pragma unroll
  for (int r = 0; r < 8; ++r) {
    const int q = r + 8 * hi;
#pragma unroll
    for (int t = 0; t < 4; ++t) orow[q * D_ + t * 16 + row16] = o[t][r];
  }
}

extern "C" void kernel_launch(void* const* d_in, const int* in_sizes, int n_in,
                              void* d_out, int out_size, void* d_ws, size_t ws_size,
                              hipStream_t stream) {
  (void)in_sizes; (void)n_in; (void)out_size; (void)ws_size;
  const float* Qm = (const float*)d_in[0];
  const float* Km = (const float*)d_in[1];
  const float* Vm = (const float*)d_in[2];
  float* Om = (float*)d_out;

  float* colmax = (float*)d_ws;                  // B*LK floats
  float* colsum = colmax + (size_t)B_ * LK;      // B*LK floats (256 KB total)

  const dim3 blk(256);                           // 8 wave32 waves / block
  const dim3 grid1(B_ * (LK / 128));             // 256 blocks
  const dim3 grid2(B_ * (LQ / 128));             // 256 blocks

  col_stats_kernel<<<grid1, blk, 0, stream>>>(Qm, Km, colmax, colsum);
  attn_out_kernel <<<grid2, blk, 0, stream>>>(Qm, Km, Vm, colmax, colsum, Om);
}